// DMIL_76725295775835
// MI455X (gfx1250) — compile-verified
//
#include <hip/hip_runtime.h>
#include <hip/hip_bf16.h>
#include <stdint.h>

#define N_PROP   500000
#define NCLS     20
#define NTOT     21
#define FG_TH    0.5f
#define BG_TH    0.1f
#define EPS_F    1e-9f
#define ABLK     512     // blocks for argmax stage 1
#define NBLK_B   512     // blocks for assign_loss (multiple of 64 for WMMA sum)

typedef __attribute__((ext_vector_type(2))) float v2f;
typedef __attribute__((ext_vector_type(8))) float v8f;

__device__ __forceinline__ float clipp(float p) {
    // matches jnp.clip(p, 1e-9, 1.0 - 1e-9) in float32 (upper bound rounds to 1.0f)
    return fminf(fmaxf(p, EPS_F), 1.0f - EPS_F);
}

// ---------------- Phase 1: per-class argmax with suppression ----------------

__global__ void argmax_stage1(const float* __restrict__ cls_prob,
                              const int* __restrict__ gt_idx,
                              const int* __restrict__ im_labels,
                              int c,
                              float* __restrict__ part_score,
                              int* __restrict__ part_idx) {
    // exclusion list: previously chosen rows for classes with positive label
    int ex[NCLS];
#pragma unroll
    for (int j = 0; j < NCLS; ++j)
        ex[j] = (j < c && im_labels[j] != 0) ? gt_idx[j] : -1;

    float best = -1.0f;
    int bidx = 0x7fffffff;
    int tid = blockIdx.x * blockDim.x + threadIdx.x;
    int stride = gridDim.x * blockDim.x;
    for (int i = tid; i < N_PROP; i += stride) {
        float p = clipp(cls_prob[(size_t)i * NTOT + 1 + c]);
        bool excl = false;
#pragma unroll
        for (int j = 0; j < NCLS; ++j) excl |= (ex[j] == i);
        // first-occurrence argmax semantics: strictly greater, or equal with lower index
        if (!excl && (p > best || (p == best && i < bidx))) { best = p; bidx = i; }
    }
    __shared__ float ss[256];
    __shared__ int   si[256];
    ss[threadIdx.x] = best; si[threadIdx.x] = bidx;
    __syncthreads();
    for (int off = 128; off > 0; off >>= 1) {
        if ((int)threadIdx.x < off) {
            float s2 = ss[threadIdx.x + off]; int i2 = si[threadIdx.x + off];
            if (s2 > ss[threadIdx.x] || (s2 == ss[threadIdx.x] && i2 < si[threadIdx.x])) {
                ss[threadIdx.x] = s2; si[threadIdx.x] = i2;
            }
        }
        __syncthreads();
    }
    if (threadIdx.x == 0) { part_score[blockIdx.x] = ss[0]; part_idx[blockIdx.x] = si[0]; }
}

__global__ void argmax_stage2(const float* __restrict__ part_score,
                              const int* __restrict__ part_idx,
                              const float* __restrict__ boxes,
                              int c,
                              int* __restrict__ gt_idx,
                              float* __restrict__ gt_score,
                              float* __restrict__ gt_boxes) {
    float best = -1.0f;
    int bidx = 0x7fffffff;
    for (int i = threadIdx.x; i < ABLK; i += blockDim.x) {
        float s = part_score[i]; int w = part_idx[i];
        if (s > best || (s == best && w < bidx)) { best = s; bidx = w; }
    }
    __shared__ float ss[256];
    __shared__ int   si[256];
    ss[threadIdx.x] = best; si[threadIdx.x] = bidx;
    __syncthreads();
    for (int off = 128; off > 0; off >>= 1) {
        if ((int)threadIdx.x < off) {
            float s2 = ss[threadIdx.x + off]; int i2 = si[threadIdx.x + off];
            if (s2 > ss[threadIdx.x] || (s2 == ss[threadIdx.x] && i2 < si[threadIdx.x])) {
                ss[threadIdx.x] = s2; si[threadIdx.x] = i2;
            }
        }
        __syncthreads();
    }
    if (threadIdx.x == 0) {
        int w = si[0];
        gt_idx[c]   = w;
        gt_score[c] = ss[0];   // already clipped
        gt_boxes[c * 4 + 0] = boxes[(size_t)w * 4 + 0];
        gt_boxes[c * 4 + 1] = boxes[(size_t)w * 4 + 1];
        gt_boxes[c * 4 + 2] = boxes[(size_t)w * 4 + 2];
        gt_boxes[c * 4 + 3] = boxes[(size_t)w * 4 + 3];
    }
}

// ---------------- Phase 2: per-proposal assignment + accumulation ----------------

__global__ void assign_loss(const float* __restrict__ cls_prob,
                            const float* __restrict__ boxes,
                            const int* __restrict__ im_labels,
                            const float* __restrict__ gt_boxes_ws,
                            const float* __restrict__ gt_score_ws,
                            float* __restrict__ neg_part,
                            float* __restrict__ sum_part,
                            int* __restrict__ cnt_part) {
    __shared__ float s_gt[NCLS * 4];
    __shared__ float s_area[NCLS];
    __shared__ float s_sc[NCLS];
    __shared__ int   s_lbl[NCLS];
    __shared__ float s_sum[NCLS];
    __shared__ int   s_cnt[NCLS];
    __shared__ float s_neg[256];

    int t = threadIdx.x;
    if (t < NCLS) {
        float gx1 = gt_boxes_ws[t * 4 + 0];
        float gy1 = gt_boxes_ws[t * 4 + 1];
        float gx2 = gt_boxes_ws[t * 4 + 2];
        float gy2 = gt_boxes_ws[t * 4 + 3];
        s_gt[t * 4 + 0] = gx1; s_gt[t * 4 + 1] = gy1;
        s_gt[t * 4 + 2] = gx2; s_gt[t * 4 + 3] = gy2;
        s_area[t] = (gx2 - gx1 + 1.0f) * (gy2 - gy1 + 1.0f);
        s_sc[t]   = gt_score_ws[t];
        s_lbl[t]  = im_labels[t];
        s_sum[t]  = 0.0f;
        s_cnt[t]  = 0;
    }
    __syncthreads();

    float neg_local = 0.0f;
    int tid = blockIdx.x * blockDim.x + threadIdx.x;
    int stride = gridDim.x * blockDim.x;
    for (int i = tid; i < N_PROP; i += stride) {
        float bx1 = boxes[(size_t)i * 4 + 0];
        float by1 = boxes[(size_t)i * 4 + 1];
        float bx2 = boxes[(size_t)i * 4 + 2];
        float by2 = boxes[(size_t)i * 4 + 3];
        float ab  = (bx2 - bx1 + 1.0f) * (by2 - by1 + 1.0f);

        float best_ov = -1.0f;
        int g = 0;
#pragma unroll
        for (int cc = 0; cc < NCLS; ++cc) {
            float ov = -1.0f;
            if (s_lbl[cc] > 0) {
                float ix1 = fmaxf(bx1, s_gt[cc * 4 + 0]);
                float iy1 = fmaxf(by1, s_gt[cc * 4 + 1]);
                float ix2 = fminf(bx2, s_gt[cc * 4 + 2]);
                float iy2 = fminf(by2, s_gt[cc * 4 + 3]);
                float iw = fmaxf(ix2 - ix1 + 1.0f, 0.0f);
                float ih = fmaxf(iy2 - iy1 + 1.0f, 0.0f);
                float inter = iw * ih;
                ov = inter / (ab + s_area[cc] - inter);
            }
            if (ov > best_ov) { best_ov = ov; g = cc; }  // first-max on ties
        }
        float w = s_sc[g];  // cls_loss_weight from RAW argmax (before bg masking)
        if (best_ov >= FG_TH) {
            float p = clipp(cls_prob[(size_t)i * NTOT + 1 + g]);
            atomicAdd(&s_sum[g], p);
            atomicAdd(&s_cnt[g], 1);
        } else if (best_ov >= BG_TH) {
            float p0 = clipp(cls_prob[(size_t)i * NTOT]);
            neg_local += -logf(p0) * w;
        }
    }

    s_neg[t] = neg_local;
    __syncthreads();
    for (int off = 128; off > 0; off >>= 1) {
        if (t < off) s_neg[t] += s_neg[t + off];
        __syncthreads();
    }
    if (t < NCLS) {
        sum_part[(size_t)blockIdx.x * NCLS + t] = s_sum[t];
        cnt_part[(size_t)blockIdx.x * NCLS + t] = s_cnt[t];
    }
    if (t == 0) neg_part[blockIdx.x] = s_neg[0];
}

// ---------------- Phase 3: finalize (single wave32, WMMA ones-matvec sum) ----------------

__global__ void finalize(const float* __restrict__ neg_part,
                         const float* __restrict__ sum_part,
                         const int* __restrict__ cnt_part,
                         const float* __restrict__ gt_score,
                         const int* __restrict__ im_labels,
                         float* __restrict__ out) {
    int lane = threadIdx.x;   // 0..31, all active (no divergence before WMMA loop)

    // Sum the NBLK_B neg partials with V_WMMA_F32_16X16X4_F32, A = ones:
    // D[m,n] = sum_k B[k,n] (+C), identical rows. Total = sum over all B operand
    // registers, which is layout-independent (64 unique values per WMMA).
    float negsum;
#if __has_builtin(__builtin_amdgcn_wmma_f32_16x16x4_f32)
    v2f a; a[0] = 1.0f; a[1] = 1.0f;
    v8f acc = {};
    for (int it = 0; it < NBLK_B / 64; ++it) {
        int base = it * 64 + lane * 2;
        v2f b; b[0] = neg_part[base]; b[1] = neg_part[base + 1];
        acc = __builtin_amdgcn_wmma_f32_16x16x4_f32(
            /*neg_a=*/false, a, /*neg_b=*/false, b,
            /*c_mod=*/(short)0, acc, /*reuse_a=*/false, /*reuse_b=*/false);
    }
    // VGPR0: lanes 0-15 -> D[0, n], lanes 16-31 -> D[8, n]; rows identical,
    // so the 32-lane sum of acc[0] is exactly 2x the column-sum total.
    negsum = acc[0];
#pragma unroll
    for (int off = 16; off > 0; off >>= 1) negsum += __shfl_xor(negsum, off, 32);
    negsum *= 0.5f;
#else
    float partial = 0.0f;
    for (int i = lane; i < NBLK_B; i += 32) partial += neg_part[i];
#pragma unroll
    for (int off = 16; off > 0; off >>= 1) partial += __shfl_xor(partial, off, 32);
    negsum = partial;
#endif

    // Per-class deterministic (fixed-order) reduction of sum/count partials.
    float s = 0.0f;
    int cnt = 0;
    if (lane < NCLS) {
        for (int b = 0; b < NBLK_B; ++b) {
            s   += sum_part[(size_t)b * NCLS + lane];
            cnt += cnt_part[(size_t)b * NCLS + lane];
        }
    }
    float pos = 0.0f;
    if (lane < NCLS && cnt > 0 && im_labels[lane] > 0) {
        float mean = s / (float)cnt;
        pos = -logf(mean) * (float)cnt * gt_score[lane];
    }
#pragma unroll
    for (int off = 16; off > 0; off >>= 1) pos += __shfl_xor(pos, off, 32);

    if (lane == 0) out[0] = (pos + negsum) / (float)N_PROP;
}

// ---------------- Host launcher ----------------

extern "C" void kernel_launch(void* const* d_in, const int* in_sizes, int n_in,
                              void* d_out, int out_size, void* d_ws, size_t ws_size,
                              hipStream_t stream) {
    (void)in_sizes; (void)n_in; (void)out_size; (void)ws_size;

    const float* cls_prob  = (const float*)d_in[0];   // [N, 21]
    const float* boxes     = (const float*)d_in[1];   // [N, 4]
    const int*   im_labels = (const int*)d_in[2];     // [1, 20]
    float*       out       = (float*)d_out;

    uint8_t* w = (uint8_t*)d_ws;
    int*   gt_idx     = (int*)  (w + 0);       // 20 ints        (pad to 128 B)
    float* gt_score   = (float*)(w + 128);     // 20 floats      (pad to 128 B)
    float* gt_boxes   = (float*)(w + 256);     // 80 floats      (pad to 512 B)
    float* part_score = (float*)(w + 768);     // ABLK floats    = 2048 B
    int*   part_idx   = (int*)  (w + 2816);    // ABLK ints      = 2048 B
    float* neg_part   = (float*)(w + 4864);    // NBLK_B floats  = 2048 B
    float* sum_part   = (float*)(w + 6912);    // NBLK_B*20 f    = 40960 B
    int*   cnt_part   = (int*)  (w + 47872);   // NBLK_B*20 i    = 40960 B
    // total workspace use: ~88832 bytes

    // Phase 1: sequential per-class argmax with suppression (cls_prob L2-resident after pass 0)
    for (int c = 0; c < NCLS; ++c) {
        argmax_stage1<<<ABLK, 256, 0, stream>>>(cls_prob, gt_idx, im_labels, c,
                                                part_score, part_idx);
        argmax_stage2<<<1, 256, 0, stream>>>(part_score, part_idx, boxes, c,
                                             gt_idx, gt_score, gt_boxes);
    }

    // Phase 2: IoU assignment + per-block partial accumulation
    assign_loss<<<NBLK_B, 256, 0, stream>>>(cls_prob, boxes, im_labels,
                                            gt_boxes, gt_score,
                                            neg_part, sum_part, cnt_part);

    // Phase 3: single-wave finalize with WMMA reduction
    finalize<<<1, 32, 0, stream>>>(neg_part, sum_part, cnt_part,
                                   gt_score, im_labels, out);
}